// ScatterRouter_38809324487112
// MI455X (gfx1250) — compile-verified
//
#include <hip/hip_runtime.h>

#define TPB 256

// ---------------------------------------------------------------------------
// Kernel A: per-token top-2 of 8 gates (+ per-block per-expert histograms)
// ---------------------------------------------------------------------------
__global__ void topk_count_kernel(const float* __restrict__ gates,
                                  int* __restrict__ top_packed,
                                  int* __restrict__ blockCounts) {
    __shared__ int cnt[8];
    const int tid = threadIdx.x;
    if (tid < 8) cnt[tid] = 0;
    __syncthreads();

    const int t = blockIdx.x * TPB + tid;
    const float4 a = ((const float4*)gates)[t * 2 + 0];
    const float4 b = ((const float4*)gates)[t * 2 + 1];
    float v[8] = {a.x, a.y, a.z, a.w, b.x, b.y, b.z, b.w};

    // argmax with ties broken toward the lowest index (matches jax.lax.top_k)
    int i0 = 0; float best = v[0];
#pragma unroll
    for (int e = 1; e < 8; ++e) { if (v[e] > best) { best = v[e]; i0 = e; } }
    int i1 = -1; float best2 = -__builtin_inff();
#pragma unroll
    for (int e = 0; e < 8; ++e) { if (e != i0 && v[e] > best2) { best2 = v[e]; i1 = e; } }

    top_packed[t] = i0 | (i1 << 8);
    atomicAdd(&cnt[i0], 1);   // order-independent -> deterministic result
    atomicAdd(&cnt[i1], 1);
    __syncthreads();
    if (tid < 8) blockCounts[blockIdx.x * 8 + tid] = cnt[tid];
}

// ---------------------------------------------------------------------------
// Kernel B (1 block / 1 wave): cross-block exclusive scan per expert,
// expert base offsets, and the `counts` output.
// ---------------------------------------------------------------------------
__global__ void scan_kernel(const int* __restrict__ blockCounts,
                            int* __restrict__ blockOffsets,
                            int* __restrict__ expertBase,
                            float* __restrict__ out_counts,
                            int NB) {
    __shared__ int totals[8];
    const int e = threadIdx.x;
    if (e < 8) {
        int run = 0;
        for (int b = 0; b < NB; ++b) {
            blockOffsets[b * 8 + e] = run;
            run += blockCounts[b * 8 + e];
        }
        totals[e]     = run;
        out_counts[e] = (float)run;
    }
    __syncthreads();
    if (e == 0) {
        int run = 0;
        for (int i = 0; i < 8; ++i) { expertBase[i] = run; run += totals[i]; }
    }
}

// ---------------------------------------------------------------------------
// Kernel C: stable position assignment. Thread e of each block walks the
// block's 256 tokens in ascending order for expert e, so the global order is
// expert-major / token-ascending — exactly the stable argsort of the mask.
// ---------------------------------------------------------------------------
__global__ void pos_kernel(const int* __restrict__ top_packed,
                           const int* __restrict__ blockOffsets,
                           const int* __restrict__ expertBase,
                           int* __restrict__ sel_token,
                           float* __restrict__ out_tags) {
    __shared__ int packed[TPB];
    const int tid    = threadIdx.x;
    const int base_t = blockIdx.x * TPB;
    packed[tid] = top_packed[base_t + tid];
    __syncthreads();
    if (tid < 8) {
        const int e = tid;
        int run = expertBase[e] + blockOffsets[blockIdx.x * 8 + e];
        for (int i = 0; i < TPB; ++i) {
            const int p  = packed[i];
            const int i0 = p & 0xff;
            const int i1 = (p >> 8) & 0xff;
            if (i0 == e || i1 == e) {
                sel_token[run] = base_t + i;
                out_tags[run]  = (float)(base_t + i);
                ++run;
            }
        }
    }
}

// ---------------------------------------------------------------------------
// Kernel D: the bandwidth stage. One block per output row; stream the 8 KB
// row through LDS with CDNA5 async global<->LDS 128-bit transfers
// (ASYNCcnt-tracked, EXEC is all ones here as required).
// LDS operand = low 32 bits of the flat shared pointer (aperture in 63:32).
// ---------------------------------------------------------------------------
__global__ void gather_kernel(const float* __restrict__ in_flow,
                              const int* __restrict__ sel_token,
                              float* __restrict__ out_data,
                              int D) {
    __shared__ char smem[8192];
    const int row = blockIdx.x;
    const int tok = sel_token[row];          // uniform across the block
    const char* src = (const char*)(in_flow  + (size_t)tok * D);
    char*       dst = (char*)(out_data + (size_t)row * D);
    const int tid = threadIdx.x;

    const unsigned lds0 = (unsigned)(unsigned long long)(void*)&smem[tid * 16];
    const unsigned lds1 = lds0 + 4096u;

    const int rowBytes = D * 4;
    for (int off = 0; off < rowBytes; off += 8192) {
        unsigned long long ga0 = (unsigned long long)(src + off + tid * 16);
        unsigned long long ga1 = ga0 + 4096ull;
        asm volatile("global_load_async_to_lds_b128 %0, %1, off"
                     :: "v"(lds0), "v"(ga0) : "memory");
        asm volatile("global_load_async_to_lds_b128 %0, %1, off"
                     :: "v"(lds1), "v"(ga1) : "memory");
        asm volatile("s_wait_asynccnt 0" ::: "memory");

        unsigned long long gd0 = (unsigned long long)(dst + off + tid * 16);
        unsigned long long gd1 = gd0 + 4096ull;
        asm volatile("global_store_async_from_lds_b128 %0, %1, off"
                     :: "v"(gd0), "v"(lds0) : "memory");
        asm volatile("global_store_async_from_lds_b128 %0, %1, off"
                     :: "v"(gd1), "v"(lds1) : "memory");
        asm volatile("s_wait_asynccnt 0" ::: "memory");
    }
}

// ---------------------------------------------------------------------------
extern "C" void kernel_launch(void* const* d_in, const int* in_sizes, int n_in,
                              void* d_out, int out_size, void* d_ws, size_t ws_size,
                              hipStream_t stream) {
    const float* in_flow = (const float*)d_in[0];
    const float* gates   = (const float*)d_in[1];
    float*       out     = (float*)d_out;

    const int N  = in_sizes[1] / 8;     // 8192 tokens
    const int D  = in_sizes[0] / N;     // 2048 features
    const int NB = N / TPB;             // 32 token blocks
    const int M  = N * 2;               // K = 2 -> 16384 routed rows

    // workspace layout (ints): top_packed[N] | blockCounts[NB*8] |
    // blockOffsets[NB*8] | expertBase[8] | sel_token[M]   (~100 KB total)
    int* ws           = (int*)d_ws;
    int* top_packed   = ws;
    int* blockCounts  = top_packed + N;
    int* blockOffsets = blockCounts + NB * 8;
    int* expertBase   = blockOffsets + NB * 8;
    int* sel_token    = expertBase + 8;

    float* out_data   = out;                        // M * D floats
    float* out_tags   = out + (size_t)M * D;        // M floats
    float* out_counts = out_tags + M;               // 8 floats

    topk_count_kernel<<<NB, TPB, 0, stream>>>(gates, top_packed, blockCounts);
    scan_kernel      <<<1, 32, 0, stream>>>(blockCounts, blockOffsets,
                                            expertBase, out_counts, NB);
    pos_kernel       <<<NB, TPB, 0, stream>>>(top_packed, blockOffsets,
                                              expertBase, sel_token, out_tags);
    gather_kernel    <<<M, TPB, 0, stream>>>(in_flow, sel_token, out_data, D);
}